// FP32ToFP16Converter_23407571764188
// MI455X (gfx1250) — compile-verified
//
#include <hip/hip_runtime.h>
#include <cstdint>

// ---------------------------------------------------------------------------
// FP32->FP16 bit-vector converter (exact integer re-derivation of the soft-
// logic reference). Streaming, memory-bound: 384 MB total -> ~16.5 us floor
// at 23.3 TB/s. Data path: HBM --(global_load_async_to_lds_b128, ASYNCcnt
// double-buffered)--> LDS --(ds_load_b128, chunk-major layout, <=2-way bank
// conflict)--> VGPR pack/convert/unpack --> non-temporal b128 stores.
// ---------------------------------------------------------------------------

typedef float v4f __attribute__((ext_vector_type(4)));

#define THREADS      256
#define BUF_BYTES    (THREADS * 128)   // 32 KB per buffer (one element/lane)
#define CHUNK_STRIDE (THREADS * 16)    // 4 KB between 16B-chunk planes

__device__ __forceinline__ void async_g2l_b128(uint32_t lds_byte_addr,
                                               const void* gaddr) {
  // GV mode: VDST = LDS byte address VGPR, VADDR = 64-bit global addr pair.
  asm volatile("global_load_async_to_lds_b128 %0, %1, off"
               :: "v"(lds_byte_addr), "v"(gaddr)
               : "memory");
}

__device__ __forceinline__ void wait_async_le8() {
#if __has_builtin(__builtin_amdgcn_s_wait_asynccnt)
  __builtin_amdgcn_s_wait_asynccnt(8);
#else
  asm volatile("s_wait_asynccnt 0x8" ::: "memory");
#endif
}
__device__ __forceinline__ void wait_async_le0() {
#if __has_builtin(__builtin_amdgcn_s_wait_asynccnt)
  __builtin_amdgcn_s_wait_asynccnt(0);
#else
  asm volatile("s_wait_asynccnt 0x0" ::: "memory");
#endif
}

__global__ __launch_bounds__(THREADS) void
fp32_to_fp16_bits_kernel(const float* __restrict__ in, float* __restrict__ out,
                         int n_elems, int n_iters) {
  __shared__ __align__(16) unsigned char lds[2 * BUF_BYTES];  // 64 KB

  const int t = threadIdx.x;
  const uint32_t lds_base = (uint32_t)(uintptr_t)&lds[0];  // LDS byte offset

  auto stage = [&](int b, int it) {
    long long e = (long long)it * THREADS + t;
    if (e >= n_elems) e = n_elems - 1;              // keep all lanes uniform
    const unsigned char* g = (const unsigned char*)(in + e * 32ll);
    const uint32_t l = lds_base + (uint32_t)b * BUF_BYTES + (uint32_t)t * 16u;
#pragma unroll
    for (int k = 0; k < 8; ++k)
      async_g2l_b128(l + (uint32_t)k * CHUNK_STRIDE, g + (size_t)k * 16);
  };

  int it = blockIdx.x;
  const int stride = gridDim.x;
  if (it >= n_iters) return;

  stage(0, it);
  int buf = 0;
  for (; it < n_iters; it += stride) {
    const int  next      = it + stride;
    const bool have_next = next < n_iters;          // block-uniform
    if (have_next) { stage(buf ^ 1, next); wait_async_le8(); }
    else           { wait_async_le0(); }

    // ---- pack 32 bit-floats -> IEEE-754 binary32 word --------------------
    const unsigned char* lbuf = &lds[(size_t)buf * BUF_BYTES + (size_t)t * 16];
    uint32_t w = 0;
#pragma unroll
    for (int k = 0; k < 8; ++k) {
      v4f f = *(const v4f*)(lbuf + (size_t)k * CHUNK_STRIDE);
      const int j = 4 * k;  // input float index of f.x (MSB-first layout)
      w |= (uint32_t)f.x << (31 - j);
      w |= (uint32_t)f.y << (30 - j);
      w |= (uint32_t)f.z << (29 - j);
      w |= (uint32_t)f.w << (28 - j);
    }

    // ---- fp32 -> fp16 (RNE), exact reference semantics -------------------
    // mux priority in reference: nan > inf > underflow > overflow > normal
    const uint32_t s  = w >> 31;
    const uint32_t e8 = (w >> 23) & 0xFFu;
    const uint32_t m  = w & 0x7FFFFFu;
    uint32_t h;
    if (e8 == 0xFFu) {
      h = (s << 15) | 0x7C00u | (m ? 0x200u : 0u);  // Inf / qNaN(mant=1000000000)
    } else if (e8 < 113u) {
      h = 0u;                                       // underflow -> +0 (sign dropped)
    } else if (e8 > 142u) {
      h = (s << 15) | 0x7C00u;                      // overflow -> Inf
    } else {
      const uint32_t mant10 = m >> 13;
      const uint32_t rbit   = (m >> 12) & 1u;
      const uint32_t sticky = (m & 0xFFFu) ? 1u : 0u;
      const uint32_t mant11 = mant10 + (rbit & (sticky | (mant10 & 1u)));
      const uint32_t exp5   = (e8 - 112u + (mant11 >> 10)) & 0x1Fu;  // 5-bit wrap
      h = (s << 15) | (exp5 << 10) | (mant11 & 0x3FFu);
    }

    // ---- unpack 16 bits -> 16 floats, streaming (non-temporal) stores ----
    const long long e = (long long)it * THREADS + t;
    if (e < n_elems) {
      v4f* o = (v4f*)(out + e * 16ll);
#pragma unroll
      for (int k = 0; k < 4; ++k) {
        const int j = 15 - 4 * k;
        v4f v;
        v.x = (float)((h >> j) & 1u);
        v.y = (float)((h >> (j - 1)) & 1u);
        v.z = (float)((h >> (j - 2)) & 1u);
        v.w = (float)((h >> (j - 3)) & 1u);
        __builtin_nontemporal_store(v, o + k);
      }
    }
    buf ^= 1;
  }
}

extern "C" void kernel_launch(void* const* d_in, const int* in_sizes, int n_in,
                              void* d_out, int out_size, void* d_ws, size_t ws_size,
                              hipStream_t stream) {
  (void)n_in; (void)out_size; (void)d_ws; (void)ws_size;
  const float* in  = (const float*)d_in[0];
  float*       out = (float*)d_out;

  const int n_elems = in_sizes[0] / 32;                 // 2048*1024 = 2,097,152
  if (n_elems <= 0) return;
  const int n_iters = (n_elems + THREADS - 1) / THREADS;  // 8192
  int grid = n_iters < 2048 ? n_iters : 2048;             // 4 iters/block: real
                                                          // double-buffer depth
  fp32_to_fp16_bits_kernel<<<grid, THREADS, 0, stream>>>(in, out, n_elems,
                                                         n_iters);
}